// SAGEReranker_14405320311458
// MI455X (gfx1250) — compile-verified
//
#include <hip/hip_runtime.h>
#include <hip/hip_bf16.h>

#define H 128
#define LDSS 130        // padded LDS row stride (floats) -> 2-bank skew per row
#define TILES 3         // 16-row tiles per block (48 rows/block)
#define ORD_NEG_INF ((int)0x807fffff)  // ordered-int encoding of -infinity

typedef __attribute__((ext_vector_type(2))) float v2f;
typedef __attribute__((ext_vector_type(8))) float v8f;

__device__ __forceinline__ int f2ord(float f) {
    int i = __float_as_int(f);
    return (i >= 0) ? i : (i ^ 0x7fffffff);
}
__device__ __forceinline__ float ord2f(int i) {
    int j = (i >= 0) ? i : (i ^ 0x7fffffff);
    return __int_as_float(j);
}

// ---------------- init ordered-int max buffer to -inf ----------------
__global__ __launch_bounds__(256) void init_m_kernel(int* __restrict__ m, int n) {
    int i = blockIdx.x * blockDim.x + threadIdx.x;
    if (i < n) m[i] = ORD_NEG_INF;
}

// ---------------- scatter-max: one wave per edge ----------------
__global__ __launch_bounds__(256) void scatter_max_kernel(
    const float* __restrict__ x, const int* __restrict__ srcIdx,
    const int* __restrict__ dstIdx, int* __restrict__ m, int E)
{
    int e    = (blockIdx.x * blockDim.x + threadIdx.x) >> 5;
    int lane = threadIdx.x & 31;
    if (e >= E) return;
    int s = srcIdx[e];
    int d = dstIdx[e];
    const float4 v = *(const float4*)(x + (size_t)s * H + lane * 4);
    int* mp = m + (size_t)d * H + lane * 4;
    atomicMax(mp + 0, f2ord(v.x));
    atomicMax(mp + 1, f2ord(v.y));
    atomicMax(mp + 2, f2ord(v.z));
    atomicMax(mp + 3, f2ord(v.w));
}

// ---------------- fused dual-GEMM + bias + (ReLU) ----------------
// out[r, j] = act( A1[r,:] @ W1[j,:] + (A2 ? A2[r,:] @ W2[j,:] : 0) + bias[j] )
// A1 is either plain floats (A1f) or ordered-int max results (A1i, decoded
// with the -inf -> 0 rule). TILES 16-row tiles per block; 8 waves cover all
// 128 output columns; each W fragment is reused across TILES accumulators.
__global__ __launch_bounds__(256) void gemm_bias_act_kernel(
    const float* __restrict__ A1f, const int* __restrict__ A1i,
    const float* __restrict__ W1,
    const float* __restrict__ A2,  const float* __restrict__ W2,
    const float* __restrict__ bias, float* __restrict__ out,
    int Nrows, int doRelu)
{
    __shared__ float sA1[TILES * 16 * LDSS];
    __shared__ float sA2[TILES * 16 * LDSS];

    const int rowBase = blockIdx.x * (TILES * 16);
    const int t = threadIdx.x;

    // Cooperative stage of the (TILES*16) x H A-tiles into LDS.
    for (int f = t; f < TILES * 16 * H; f += 256) {
        int r = f >> 7;          // / H   (0 .. TILES*16-1)
        int c = f & (H - 1);     // % H
        int row = rowBase + r;
        float v1 = 0.0f, v2 = 0.0f;
        if (row < Nrows) {
            size_t g = (size_t)row * H + c;
            if (A1i) {
                int iv = A1i[g];
                v1 = (iv == ORD_NEG_INF) ? 0.0f : ord2f(iv);  // isolated -> 0
            } else {
                v1 = A1f[g];
            }
            if (A2) v2 = A2[g];
        }
        sA1[r * LDSS + c] = v1;
        if (A2) sA2[r * LDSS + c] = v2;
    }
    __syncthreads();

    const int lane = t & 31;
    const int wave = t >> 5;
    const int nIdx = lane & 15;                 // N / M-row index within tile
    const int koff = (lane < 16) ? 0 : 2;       // K sub-offset per half-wave
    const int col  = wave * 16 + nIdx;          // output column (0..127)
    const int aOff = nIdx * LDSS + koff;        // per-lane A base within a tile

    v8f acc[TILES];
    #pragma unroll
    for (int tt = 0; tt < TILES; ++tt)
        acc[tt] = (v8f){0.f, 0.f, 0.f, 0.f, 0.f, 0.f, 0.f, 0.f};

    // GEMM 1: A1 @ W1^T  (B[k][j] = W1[j*H + k] -> contiguous float2 in k).
    // One W fragment feeds TILES WMMAs.
    #pragma unroll
    for (int kb = 0; kb < H; kb += 4) {
        v2f b = *(const v2f*)(&W1[(size_t)col * H + kb + koff]);
        #pragma unroll
        for (int tt = 0; tt < TILES; ++tt) {
            v2f a = *(const v2f*)(&sA1[tt * (16 * LDSS) + aOff + kb]);
            acc[tt] = __builtin_amdgcn_wmma_f32_16x16x4_f32(
                false, a, false, b, (short)0, acc[tt], false, false);
        }
    }
    // GEMM 2: A2 @ W2^T (accumulates into same C).
    if (A2) {
        #pragma unroll
        for (int kb = 0; kb < H; kb += 4) {
            v2f b = *(const v2f*)(&W2[(size_t)col * H + kb + koff]);
            #pragma unroll
            for (int tt = 0; tt < TILES; ++tt) {
                v2f a = *(const v2f*)(&sA2[tt * (16 * LDSS) + aOff + kb]);
                acc[tt] = __builtin_amdgcn_wmma_f32_16x16x4_f32(
                    false, a, false, b, (short)0, acc[tt], false, false);
            }
        }
    }

    // Epilogue: bias + optional ReLU, store per C/D VGPR layout (row-guarded).
    const float bcol  = bias[col];
    const int   rbase = (lane < 16) ? 0 : 8;
    #pragma unroll
    for (int tt = 0; tt < TILES; ++tt) {
        #pragma unroll
        for (int i = 0; i < 8; ++i) {
            int row = rowBase + tt * 16 + rbase + i;
            if (row < Nrows) {
                float v = acc[tt][i] + bcol;
                if (doRelu) v = fmaxf(v, 0.0f);
                out[(size_t)row * H + col] = v;
            }
        }
    }
}

// ---------------- cosine score: one wave per node ----------------
__global__ __launch_bounds__(256) void score_kernel(
    const float* __restrict__ x, const float* __restrict__ q,
    const int* __restrict__ batch, float* __restrict__ out, int N)
{
    int node = (blockIdx.x * blockDim.x + threadIdx.x) >> 5;
    int lane = threadIdx.x & 31;
    if (node >= N) return;
    int g = batch[node];
    const float4 xv = *(const float4*)(x + (size_t)node * H + lane * 4);
    const float4 qv = *(const float4*)(q + (size_t)g * H + lane * 4);
    float dot = xv.x * qv.x + xv.y * qv.y + xv.z * qv.z + xv.w * qv.w;
    float xx  = xv.x * xv.x + xv.y * xv.y + xv.z * xv.z + xv.w * xv.w;
    float qq  = qv.x * qv.x + qv.y * qv.y + qv.z * qv.z + qv.w * qv.w;
    #pragma unroll
    for (int off = 16; off > 0; off >>= 1) {
        dot += __shfl_xor(dot, off, 32);
        xx  += __shfl_xor(xx,  off, 32);
        qq  += __shfl_xor(qq,  off, 32);
    }
    if (lane == 0) {
        float nx = fmaxf(sqrtf(xx), 1e-12f);
        float nq = fmaxf(sqrtf(qq), 1e-12f);
        out[node] = dot / (nx * nq);
    }
}

extern "C" void kernel_launch(void* const* d_in, const int* in_sizes, int n_in,
                              void* d_out, int out_size, void* d_ws, size_t ws_size,
                              hipStream_t stream) {
    const float* x0    = (const float*)d_in[0];
    const float* query = (const float*)d_in[1];
    const float* Wl    = (const float*)d_in[2];
    const float* bl    = (const float*)d_in[3];
    const float* Wr    = (const float*)d_in[4];
    const float* Wlin  = (const float*)d_in[5];
    const float* blin  = (const float*)d_in[6];
    const int*   edge  = (const int*)d_in[7];
    const int*   batch = (const int*)d_in[8];

    const int N = in_sizes[0] / H;     // 50000
    const int E = in_sizes[7] / 2;     // 640000
    const int L = in_sizes[3] / H;     // 3
    const size_t NH = (size_t)N * H;

    // Workspace: [0, NH) ordered-int m buffer; [NH, 2NH) and [2NH, 3NH) x ping-pong.
    int*   bufM = (int*)d_ws;
    float* bufA = (float*)d_ws + NH;
    float* bufB = (float*)d_ws + 2 * NH;
    float* bufs[2] = {bufA, bufB};

    const int* src = edge;
    const int* dst = edge + E;

    const int tiles16    = (N + 15) / 16;                    // 3125
    const int gemmBlocks = (tiles16 + TILES - 1) / TILES;    // 1042
    const int initBlocks = (int)((NH + 255) / 256);
    const int smBlocks   = (int)(((size_t)E * 32 + 255) / 256);
    const int scBlocks   = (int)(((size_t)N * 32 + 255) / 256);

    const float* xin = x0;
    for (int l = 0; l < L; ++l) {
        init_m_kernel<<<initBlocks, 256, 0, stream>>>(bufM, (int)NH);
        scatter_max_kernel<<<smBlocks, 256, 0, stream>>>(xin, src, dst, bufM, E);
        float* xout = bufs[l & 1];
        gemm_bias_act_kernel<<<gemmBlocks, 256, 0, stream>>>(
            nullptr, bufM, Wl + (size_t)l * H * H,
            xin, Wr + (size_t)l * H * H,
            bl + (size_t)l * H, xout, N, 1);
        xin = xout;
    }
    // Final linear (no aggregation, no ReLU).
    float* xfin = bufs[L & 1];
    gemm_bias_act_kernel<<<gemmBlocks, 256, 0, stream>>>(
        xin, nullptr, Wlin, nullptr, nullptr, blin, xfin, N, 0);

    score_kernel<<<scBlocks, 256, 0, stream>>>(xfin, query, batch, (float*)d_out, N);
}